// DetectionLoss_49589692400138
// MI455X (gfx1250) — compile-verified
//
#include <hip/hip_runtime.h>
#include <hip/hip_bf16.h>

// ---------------- problem constants (from reference) ----------------
#define B_    16
#define N_    65536
#define C_    80
#define M_    32
#define TILE  128                 // anchors per block == threads per block
#define NBLK_PER_B (N_ / TILE)    // 512
#define NBLK       (B_ * NBLK_PER_B) // 8192
#define ROWSTRIDE  85             // 80 dwords + 1 pad dword per 16 dwords (TDM pad)

typedef unsigned int u32x4 __attribute__((ext_vector_type(4)));
typedef int          i32x8 __attribute__((ext_vector_type(8)));
typedef int          i32x4 __attribute__((ext_vector_type(4)));
typedef float        f32x2 __attribute__((ext_vector_type(2)));
typedef float        f32x8 __attribute__((ext_vector_type(8)));

// Wave32 sum via V_WMMA_F32_16X16X4_F32 with B == ones:
// A[m][0] = v(lane m), A[m][2] = v(lane m+16)  ->  D[m][n] = v_m + v_{m+16}.
// Summing the 8 D VGPRs gives lanes 0-15: sum(rows 0-7), lanes 16-31: sum(rows 8-15);
// one shfl_xor(16) completes the 32-lane total (all lanes hold it).
// Requires EXEC all-ones: only called from non-divergent, wave-uniform paths.
__device__ __forceinline__ float wave_sum(float v) {
  f32x2 a;   a[0] = v;    a[1] = 0.0f;
  f32x2 one; one[0] = 1.0f; one[1] = 1.0f;
  f32x8 c = {};
  c = __builtin_amdgcn_wmma_f32_16x16x4_f32(false, a, false, one,
                                            (short)0, c, false, false);
  float r = ((c[0] + c[1]) + (c[2] + c[3])) + ((c[4] + c[5]) + (c[6] + c[7]));
  r += __shfl_xor(r, 16, 32);
  return r;
}

__device__ __forceinline__ float block_sum(float v, float* sh) {
  v = wave_sum(v);
  const int lane = threadIdx.x & 31;
  const int wid  = threadIdx.x >> 5;
  if (lane == 0) sh[wid] = v;
  __syncthreads();
  const int nw = (int)(blockDim.x >> 5);
  float r = ((int)threadIdx.x < nw) ? sh[threadIdx.x] : 0.0f;
  if (wid == 0) r = wave_sum(r);      // wave-uniform branch: EXEC all-ones inside
  __syncthreads();
  return r;                            // valid on thread 0
}

__global__ __launch_bounds__(TILE) void det_loss_main(
    const float* __restrict__ cls,  const float* __restrict__ boxes,
    const float* __restrict__ objn, const float* __restrict__ gtb,
    const int*   __restrict__ gtl,  const int*   __restrict__ nobj,
    float* __restrict__ wfl, float* __restrict__ wbx,
    float* __restrict__ wcn, float* __restrict__ wob)
{
  __shared__ float tile[TILE * ROWSTRIDE];   // 43.5 KB, TDM-padded rows
  __shared__ float sGT[M_ * 4];
  __shared__ int   sLab[M_];
  __shared__ int   sN;
  __shared__ float sh[8];

  const int b    = blockIdx.y;
  const int base = blockIdx.x * TILE;
  const int t    = threadIdx.x;

  // ---- wave 0: kick off TDM DMA of the 128x80 f32 classification tile ----
  if (t < 32) {
    unsigned long long ga =
        (unsigned long long)(cls + ((size_t)b * N_ + base) * C_);
    unsigned lds = (unsigned)(unsigned long long)(void*)&tile[0]; // LDS byte offset
    u32x4 g0;
    g0.x = 1u;                                   // count=1 valid user descriptor
    g0.y = lds;                                  // lds_addr
    g0.z = (unsigned)(ga & 0xFFFFFFFFull);       // global_addr[31:0]
    g0.w = (unsigned)((ga >> 32) & 0x01FFFFFFull) | 0x80000000u; // [56:32] | type=2
    i32x8 g1;
    g1[0] = 0x00D20000;                 // data_size=4B, pad_enable, interval=16dw, amount=1dw
    g1[1] = (int)(80u << 16);           // tensor_dim0 = 80 elements/row
    g1[2] = (int)((unsigned)TILE << 16);// tensor_dim1 = 128 rows
    g1[3] = (int)(80u << 16);           // tile_dim0 = 80
    g1[4] = TILE;                       // tile_dim1 = 128
    g1[5] = 80;                         // tensor_dim0_stride = 80 elements
    g1[6] = 0;
    g1[7] = 0;
    i32x4 gz4 = {0, 0, 0, 0};           // 2D tensor: groups 2/3 unused
    i32x8 gz8 = {0, 0, 0, 0, 0, 0, 0, 0};
    // 6-arg form on this toolchain: (g0, g1, g2, g3, extra, cpol)
    __builtin_amdgcn_tensor_load_to_lds(g0, g1, gz4, gz4, gz8, 0);
  }

  // ---- GT data -> LDS (overlaps with in-flight TDM) ----
  if (t < M_ * 4) sGT[t]  = gtb[(size_t)b * (M_ * 4) + t];
  if (t < M_)     sLab[t] = gtl[b * M_ + t];
  if (t == 0)     sN      = nobj[b];
  __syncthreads();

  // ---- per-anchor assignment (last valid GT box containing the anchor) ----
  const int   idx = base + t;
  const float ax  = (float)(idx & 255) * (1.0f / 255.0f);
  const float ay  = (float)(idx >> 8)  * (1.0f / 255.0f);
  const int   nv  = sN;
  int jl = -1;
  #pragma unroll 8
  for (int j = 0; j < M_; ++j) {
    if (j < nv) {
      const float cx = sGT[j * 4 + 0], cy = sGT[j * 4 + 1];
      const float hw = 0.5f * sGT[j * 4 + 2], hh = 0.5f * sGT[j * 4 + 3];
      const bool inside = (ax >= cx - hw) && (ax <= cx + hw) &&
                          (ay >= cy - hh) && (ay <= cy + hh);
      if (inside) jl = j;
    }
  }
  const float mask = (jl >= 0) ? 1.0f : 0.0f;
  float acx = 0.f, acy = 0.f, aw = 0.f, ah = 0.f;
  int   pos = 0;
  if (jl >= 0) {
    acx = sGT[jl * 4 + 0]; acy = sGT[jl * 4 + 1];
    aw  = sGT[jl * 4 + 2]; ah  = sGT[jl * 4 + 3];
    pos = sLab[jl];                    // clip(label-1,0,C-1) == gt label here
  }

  // ---- IoU + objectness BCE (coalesced vector loads; hides TDM latency) ----
  const float4 pb = ((const float4*)boxes)[(size_t)b * N_ + idx];
  const float  p  = objn[(size_t)b * N_ + idx];

  const float b1x1 = pb.x - 0.5f * pb.z, b1x2 = pb.x + 0.5f * pb.z;
  const float b1y1 = pb.y - 0.5f * pb.w, b1y2 = pb.y + 0.5f * pb.w;
  const float b2x1 = acx - 0.5f * aw,    b2x2 = acx + 0.5f * aw;
  const float b2y1 = acy - 0.5f * ah,    b2y2 = acy + 0.5f * ah;
  const float iw    = fmaxf(fminf(b1x2, b2x2) - fmaxf(b1x1, b2x1), 0.0f);
  const float ih    = fmaxf(fminf(b1y2, b2y2) - fmaxf(b1y1, b2y1), 0.0f);
  const float inter = iw * ih;
  const float uni   = pb.z * pb.w + aw * ah - inter;
  const float iou   = inter / (uni + 1e-8f);

  const float objterm = mask * fmaxf(__logf(p), -100.0f) +
                        (1.0f - mask) * fmaxf(log1pf(-p), -100.0f);

  // ---- focal loss from the LDS tile (bank-conflict-free: stride 85 dwords) ----
  __builtin_amdgcn_s_wait_tensorcnt(0);
  __syncthreads();

  const float* row = &tile[t * ROWSTRIDE];
  float mx = -3.402823466e+38f;
  #pragma unroll
  for (int g = 0; g < 5; ++g)
    #pragma unroll
    for (int k = 0; k < 16; ++k)
      mx = fmaxf(mx, row[g * 17 + k]);
  float s = 0.0f;
  #pragma unroll
  for (int g = 0; g < 5; ++g)
    #pragma unroll
    for (int k = 0; k < 16; ++k)
      s += __expf(row[g * 17 + k] - mx);
  const float xp  = row[pos + (pos >> 4)];
  const float lse = mx + __logf(s);
  const float ce  = lse - xp;
  const float pt  = __expf(-ce);
  const float om  = 1.0f - pt;
  const float fl  = 0.25f * om * om * ce * mask;
  const float bx  = (1.0f - iou) * mask;

  // ---- deterministic per-block partials ----
  const float rfl = block_sum(fl,      sh);
  const float rbx = block_sum(bx,      sh);
  const float rcn = block_sum(mask,    sh);
  const float rob = block_sum(objterm, sh);
  if (t == 0) {
    const int pidx = b * NBLK_PER_B + blockIdx.x;
    wfl[pidx] = rfl; wbx[pidx] = rbx; wcn[pidx] = rcn; wob[pidx] = rob;
  }
}

__global__ __launch_bounds__(256) void det_loss_final(
    const float* __restrict__ wfl, const float* __restrict__ wbx,
    const float* __restrict__ wcn, const float* __restrict__ wob,
    float* __restrict__ out)
{
  __shared__ float sh[8];
  const int t = threadIdx.x;
  float cls_acc = 0.0f, box_acc = 0.0f;
  for (int b = 0; b < B_; ++b) {
    float f = 0.f, x = 0.f, c = 0.f;
    for (int k = t; k < NBLK_PER_B; k += 256) {
      f += wfl[b * NBLK_PER_B + k];
      x += wbx[b * NBLK_PER_B + k];
      c += wcn[b * NBLK_PER_B + k];
    }
    f = block_sum(f, sh);
    x = block_sum(x, sh);
    c = block_sum(c, sh);
    if (t == 0) {
      const float d = fmaxf(c, 1.0f);   // counts==0 => f==x==0, matches `where`
      cls_acc += f / d;
      box_acc += x / d;
    }
  }
  float o = 0.0f;
  for (int k = t; k < NBLK; k += 256) o += wob[k];
  o = block_sum(o, sh);
  if (t == 0) {
    const float cls_loss = cls_acc / (float)B_;   // total_pos==0 => acc==0, same
    const float box_loss = box_acc / (float)B_;
    const float obj_loss = -o / (float)(B_ * N_);
    out[0] = 1.0f * cls_loss + 5.0f * box_loss + 1.0f * obj_loss;
  }
}

extern "C" void kernel_launch(void* const* d_in, const int* in_sizes, int n_in,
                              void* d_out, int out_size, void* d_ws, size_t ws_size,
                              hipStream_t stream) {
  (void)in_sizes; (void)n_in; (void)out_size; (void)ws_size;
  const float* cls   = (const float*)d_in[0];
  const float* boxes = (const float*)d_in[1];
  const float* objn  = (const float*)d_in[2];
  const float* gtb   = (const float*)d_in[3];
  const int*   gtl   = (const int*)d_in[4];
  const int*   nob   = (const int*)d_in[5];

  float* ws  = (float*)d_ws;           // 4 * 8192 floats = 128 KB of partials
  float* wfl = ws;
  float* wbx = ws + NBLK;
  float* wcn = ws + 2 * NBLK;
  float* wob = ws + 3 * NBLK;

  dim3 grid(NBLK_PER_B, B_);
  det_loss_main<<<grid, TILE, 0, stream>>>(cls, boxes, objn, gtb, gtl, nob,
                                           wfl, wbx, wcn, wob);
  det_loss_final<<<1, 256, 0, stream>>>(wfl, wbx, wcn, wob, (float*)d_out);
}